// MSHGNN_72129680769631
// MI455X (gfx1250) — compile-verified
//
#include <hip/hip_runtime.h>

// Problem constants
#define BDIM  16
#define ORDER 2
#define NN    512
#define DD    128
#define HHN   8
#define DHH   16
#define NEG_SLOPE 0.2f

typedef __attribute__((ext_vector_type(2))) float v2f;
typedef __attribute__((ext_vector_type(8))) float v8f;

// Workspace layout (in floats)
//  xp   : [4][8192][128]   projected features per (order,conv)
//  ssrc : [4][8192][8]
//  sdst : [4][8192][8]
//  hbuf : [4][8192][128]   per-GAT output (combined in finalize)
#define XP_T   (8192ull * 128ull)
#define SS_T   (8192ull * 8ull)
#define XP_OFF 0ull
#define SS_OFF (4ull * XP_T)
#define SD_OFF (SS_OFF + 4ull * SS_T)
#define HB_OFF (SD_OFF + 4ull * SS_T)

// ---------------------------------------------------------------------------
// Kernel 1: xp[t] = x[:,i] @ W[t]^T + pb[t]   (t = i*2+j), f32 WMMA 16x16x4
// One wave computes one 16(M)x16(N) tile; K=128 in 32 wmma steps.
// ---------------------------------------------------------------------------
__global__ __launch_bounds__(256) void proj_kernel(
    const float* __restrict__ x, const float* __restrict__ W,
    const float* __restrict__ pb, float* __restrict__ ws) {
  int wid  = blockIdx.x * 8 + (threadIdx.x >> 5);   // global wave id
  int lane = threadIdx.x & 31;
  int m16  = lane & 15;
  int half = lane >> 4;

  int t     = wid >> 12;        // 4096 tiles per t
  int rem   = wid & 4095;
  int mtile = rem >> 3;         // 512 M-tiles
  int ntile = rem & 7;          // 8 N-tiles
  int i     = t >> 1;

  // A operand: x row (M = mtile*16 + m16), K pair at kb + 2*half
  int mrow = mtile * 16 + m16;                // 0..8191 == b*512 + n
  int b = mrow >> 9, n = mrow & 511;
  const float* xrow = x + (((size_t)b * ORDER + i) * NN + n) * DD;
  // B operand: B[k, nout] = W[t][nout][k]
  const float* wrow = W + ((size_t)t * DD + ntile * 16 + m16) * DD;

  // C init = bias (depends only on N column = m16)
  float pbv = pb[t * DD + ntile * 16 + m16];
  v8f c;
#pragma unroll
  for (int r = 0; r < 8; ++r) c[r] = pbv;

#pragma unroll 4
  for (int kb = 0; kb < DD; kb += 4) {
    int k0 = kb + 2 * half;
    v2f a, bv;
    a.x  = xrow[k0];  a.y  = xrow[k0 + 1];
    bv.x = wrow[k0];  bv.y = wrow[k0 + 1];
    c = __builtin_amdgcn_wmma_f32_16x16x4_f32(false, a, false, bv,
                                              (short)0, c, false, false);
  }

  float* xp = ws + XP_OFF + (size_t)t * XP_T;
#pragma unroll
  for (int r = 0; r < 8; ++r) {
    int mo = mtile * 16 + r + 8 * half;       // C row M = r + 8*half
    xp[(size_t)mo * DD + ntile * 16 + m16] = c[r];
  }
}

// ---------------------------------------------------------------------------
// Kernel 2: s_src/s_dst[t][m][h] = <xh[m,h,:], att_{src,dst}[t,h,:]>
// ---------------------------------------------------------------------------
__global__ void scores_kernel(const float* __restrict__ att_src,
                              const float* __restrict__ att_dst,
                              float* __restrict__ ws) {
  int idx = blockIdx.x * blockDim.x + threadIdx.x;
  if (idx >= 4 * 8192 * HHN) return;
  int t = idx >> 16;            // 8192*8 = 65536 per t
  int rem = idx & 65535;
  int m = rem >> 3;
  int h = rem & 7;
  const float* xr = ws + XP_OFF + (size_t)t * XP_T + (size_t)m * DD + h * DHH;
  const float* as = att_src + ((size_t)t * HHN + h) * DHH;
  const float* ad = att_dst + ((size_t)t * HHN + h) * DHH;
  float s1 = 0.f, s2 = 0.f;
#pragma unroll
  for (int f = 0; f < DHH; ++f) { float v = xr[f]; s1 += v * as[f]; s2 += v * ad[f]; }
  ws[SS_OFF + (size_t)t * SS_T + (size_t)m * HHN + h] = s1;
  ws[SD_OFF + (size_t)t * SS_T + (size_t)m * HHN + h] = s2;
}

// ---------------------------------------------------------------------------
// Kernel 3: fused masked-softmax attention + aggregation.
// Block = (t, b, h). Wave computes out tile [16 dst x 16 f] = E^T @ X with
// f32 WMMA 16x16x4, generating E values directly in the A-operand layout.
// Softmax denom accumulated alongside (non-edge contributes exp(0)=1).
// ---------------------------------------------------------------------------
__global__ __launch_bounds__(256) void attn_kernel(
    const int* __restrict__ A, const float* __restrict__ bias,
    float* __restrict__ ws) {
  __shared__ float Xs[NN * DHH];     // 32 KB: xh slice [src][f]
  __shared__ float ssrc_s[NN];
  __shared__ float sdst_s[NN];

  int bid = blockIdx.x;              // 512 = 4*16*8
  int t   = bid >> 7;
  int rem = bid & 127;
  int b   = rem >> 3;
  int hh  = rem & 7;
  int i   = t >> 1, j = t & 1;
  int tid = threadIdx.x;

  const float* xp = ws + XP_OFF + (size_t)t * XP_T + (size_t)b * NN * DD;
  for (int idx = tid; idx < NN * DHH; idx += 256) {
    int s = idx >> 4, f = idx & 15;
    Xs[idx] = xp[(size_t)s * DD + hh * DHH + f];
  }
  const float* ssp = ws + SS_OFF + (size_t)t * SS_T + (size_t)b * NN * HHN;
  const float* sdp = ws + SD_OFF + (size_t)t * SS_T + (size_t)b * NN * HHN;
  for (int s = tid; s < NN; s += 256) {
    ssrc_s[s] = ssp[(size_t)s * HHN + hh];
    sdst_s[s] = sdp[(size_t)s * HHN + hh];
  }
  __syncthreads();

  int lane = tid & 31, wid = tid >> 5;
  int m16 = lane & 15, half = lane >> 4;
  const int*  Ab = A + ((size_t)b * ORDER + i) * NN * NN;    // [src][dst]
  float*      hb = ws + HB_OFF + (size_t)t * XP_T + (size_t)b * NN * DD;
  const float* bsp = bias + (size_t)t * DD + hh * DHH;

  for (int dt = wid; dt < 32; dt += 8) {                     // 32 dst tiles
    int dbase = dt * 16;
    int d = dbase + m16;                                     // this lane's dst
    float sd = sdst_s[d];
    v8f c = {0.f, 0.f, 0.f, 0.f, 0.f, 0.f, 0.f, 0.f};
    float dsum = 0.f;

    for (int kb = 0; kb < NN; kb += 4) {
      int s0 = kb + 2 * half;                                // A-layout K pair
      int sp = (s0 + 64 < NN) ? (s0 + 64) : s0;
      __builtin_prefetch(&Ab[(size_t)sp * NN + d], 0, 1);    // stream A from L2

      int a0 = Ab[(size_t)s0 * NN + d];
      int a1 = Ab[(size_t)(s0 + 1) * NN + d];
      bool e0m = j == 0 ? (a0 == 2 || a0 == 4) : (a0 == 3 || a0 == 4);
      bool e1m = j == 0 ? (a1 == 2 || a1 == 4) : (a1 == 3 || a1 == 4);
      float sc0 = ssrc_s[s0] + sd;
      float sc1 = ssrc_s[s0 + 1] + sd;
      sc0 = sc0 >= 0.f ? sc0 : NEG_SLOPE * sc0;              // leaky relu
      sc1 = sc1 >= 0.f ? sc1 : NEG_SLOPE * sc1;
      float e0 = e0m ? __expf(sc0) : 1.0f;                   // masked -> exp(0)=1
      float e1 = e1m ? __expf(sc1) : 1.0f;
      dsum += e0 + e1;

      v2f av; av.x = e0; av.y = e1;                          // E^T tile (A op)
      v2f bv; bv.x = Xs[s0 * DHH + m16];                     // X tile (B op)
      bv.y = Xs[(s0 + 1) * DHH + m16];
      c = __builtin_amdgcn_wmma_f32_16x16x4_f32(false, av, false, bv,
                                                (short)0, c, false, false);
    }

    // Softmax denominator for dst = dbase + m16 (combine both lane halves)
    dsum += __shfl_xor(dsum, 16);
    float rden = 1.0f / dsum;

#pragma unroll
    for (int r = 0; r < 8; ++r) {
      int dd = dbase + r + 8 * half;                         // C row M
      float scale = __shfl(rden, r + 8 * half);              // denom[dd]
      float val = c[r] * scale
                + xp[(size_t)dd * DD + hh * DHH + m16]       // add_self_loops
                + bsp[m16];                                  // output bias
      hb[(size_t)dd * DD + hh * DHH + m16] = val;
    }
  }
}

// ---------------------------------------------------------------------------
// Kernel 4: h[b,i] = gat(i,0)+gat(i,1);  out[:,i] = h[:,i] + 0.5*(h[:,0]+h[:,1])
// ---------------------------------------------------------------------------
__global__ void finalize_kernel(const float* __restrict__ ws,
                                float* __restrict__ out) {
  int idx = blockIdx.x * blockDim.x + threadIdx.x;
  if (idx >= BDIM * ORDER * NN * DD) return;
  int b  = idx >> 17;                 // 2*512*128 per batch
  int i  = (idx >> 16) & 1;
  int nm = idx & 65535;               // n*128 + d
  size_t m = ((size_t)b << 9) | (nm >> 7);
  int dcol = nm & 127;
  const float* hbp = ws + HB_OFF;
  float h0 = hbp[0 * XP_T + m * DD + dcol] + hbp[1 * XP_T + m * DD + dcol];
  float h1 = hbp[2 * XP_T + m * DD + dcol] + hbp[3 * XP_T + m * DD + dcol];
  float hi = (i == 0) ? h0 : h1;
  out[idx] = hi + 0.5f * (h0 + h1);
}

// ---------------------------------------------------------------------------
extern "C" void kernel_launch(void* const* d_in, const int* in_sizes, int n_in,
                              void* d_out, int out_size, void* d_ws, size_t ws_size,
                              hipStream_t stream) {
  const float* x       = (const float*)d_in[0];
  const int*   A       = (const int*)d_in[1];
  const float* W       = (const float*)d_in[2];
  const float* pb      = (const float*)d_in[3];
  const float* att_src = (const float*)d_in[4];
  const float* att_dst = (const float*)d_in[5];
  const float* bias    = (const float*)d_in[6];
  float* ws  = (float*)d_ws;
  float* out = (float*)d_out;

  // 16384 wave-tiles / 8 waves per block
  proj_kernel<<<2048, 256, 0, stream>>>(x, W, pb, ws);
  scores_kernel<<<(4 * 8192 * HHN + 255) / 256, 256, 0, stream>>>(att_src, att_dst, ws);
  attn_kernel<<<4 * BDIM * HHN, 256, 0, stream>>>(A, bias, ws);
  finalize_kernel<<<(BDIM * ORDER * NN * DD + 255) / 256, 256, 0, stream>>>(ws, out);
}